// Injector_31301721653963
// MI455X (gfx1250) — compile-verified
//
#include <hip/hip_runtime.h>
#include <hip/hip_bf16.h>

// ---------------------------------------------------------------------------
// Problem constants (match reference)
#define DD     128
#define HH     2
#define NENT   100000
#define BQ     64
#define RREL   500
#define EE     400000
#define NEG    0.2f

typedef float v2f __attribute__((ext_vector_type(2)));
typedef float v8f __attribute__((ext_vector_type(8)));

// monotonic float->uint key for atomic max on signed floats
__device__ __forceinline__ unsigned fkey(float f) {
    unsigned u = __float_as_uint(f);
    return (u & 0x80000000u) ? ~u : (u | 0x80000000u);
}
__device__ __forceinline__ float fdecode(unsigned k) {
    unsigned u = (k & 0x80000000u) ? (k & 0x7FFFFFFFu) : ~k;
    return __uint_as_float(u);
}
__device__ __forceinline__ float lrelu(float x) { return x > 0.f ? x : NEG * x; }

// ---------------------------------------------------------------------------
// K0: zero the atomic-max keys, denominators, and the node-output region
__global__ void init_buffers(unsigned* amaxkey, float* denom, float* out,
                             int nkeys, int nout) {
    int i = blockIdx.x * blockDim.x + threadIdx.x;
    if (i < nkeys) { amaxkey[i] = 0u; denom[i] = 0.f; }
    if (i < nout)  out[i] = 0.f;
}

// K1: rel_feat = vstack(relations, ones(1,128))
__global__ void build_relfeat(const float* __restrict__ relations,
                              float* __restrict__ relfeat) {
    int i = blockIdx.x * blockDim.x + threadIdx.x;
    if (i < RREL * DD)              relfeat[i] = relations[i];
    else if (i < (RREL + 1) * DD)   relfeat[i] = 1.0f;
}

// ---------------------------------------------------------------------------
// K2: x_l / x_r = node @ W.T + b  via V_WMMA_F32_16X16X4_F32
// node rows: [0,NENT) -> entities, [NENT, n_tot) -> queries.
// One wave computes one 16x16 output tile; K=128 in 32 steps of 4.
// gridDim.z selects (W_l,b_l,x_l) vs (W_r,b_r,x_r).
__global__ __launch_bounds__(256)
void gemm_xlr(const float* __restrict__ ent, const float* __restrict__ qry,
              const float* __restrict__ Wl, const float* __restrict__ bl,
              const float* __restrict__ Wr, const float* __restrict__ br,
              float* __restrict__ xl, float* __restrict__ xr, int n_tot) {
    const int nTilesN = 256 / 16;                 // 16
    int wavesPerBlock = blockDim.x >> 5;
    int wid  = blockIdx.x * wavesPerBlock + (threadIdx.x >> 5);
    int lane = threadIdx.x & 31;
    int mTiles = n_tot >> 4;                      // 100064/16 = 6254 exactly
    if (wid >= mTiles * nTilesN) return;
    int mt = wid / nTilesN, nt = wid % nTilesN;

    const float* W    = (blockIdx.z == 0) ? Wl : Wr;
    const float* bias = (blockIdx.z == 0) ? bl : br;
    float*       out  = (blockIdx.z == 0) ? xl : xr;

    int half = lane >> 4;                          // K half of the step
    int l15  = lane & 15;
    int arow = mt * 16 + l15;
    const float* Ap = (arow < NENT) ? (ent + (size_t)arow * DD)
                                    : (qry + (size_t)(arow - NENT) * DD);
    const float* Wp = W + (size_t)(nt * 16 + l15) * DD;   // B[k][n] = W[n][k]
    Ap += half * 2;
    Wp += half * 2;

    v8f c = {0.f, 0.f, 0.f, 0.f, 0.f, 0.f, 0.f, 0.f};
#pragma unroll
    for (int k = 0; k < DD; k += 4) {
        v2f a = *(const v2f*)(Ap + k);
        v2f b = *(const v2f*)(Wp + k);
        c = __builtin_amdgcn_wmma_f32_16x16x4_f32(false, a, false, b,
                                                  (short)0, c, false, false);
    }
    float bv = bias[nt * 16 + l15];
    size_t rowBase = (size_t)(mt * 16 + half * 8);
#pragma unroll
    for (int r = 0; r < 8; ++r)
        out[(rowBase + r) * 256 + nt * 16 + l15] = c[r] + bv;
}

// ---------------------------------------------------------------------------
// K3/K8: generic WMMA GEMM  C[M x N] = f(A[M x K]) @ W[N x K]^T (+bias)
// relu==1 applies relu to A-fragments (for the out_edge MLP).
__global__ __launch_bounds__(256)
void gemm_generic(const float* __restrict__ A, const float* __restrict__ W,
                  const float* __restrict__ bias, float* __restrict__ C,
                  int M, int K, int N, int relu) {
    int wavesPerBlock = blockDim.x >> 5;
    int wid  = blockIdx.x * wavesPerBlock + (threadIdx.x >> 5);
    int lane = threadIdx.x & 31;
    int mTiles = (M + 15) >> 4;
    int nTiles = N >> 4;
    if (wid >= mTiles * nTiles) return;
    int mt = wid / nTiles, nt = wid % nTiles;

    int half = lane >> 4;
    int l15  = lane & 15;
    int arow = mt * 16 + l15; if (arow >= M) arow = M - 1;   // clamp (dup rows)
    const float* Ap = A + (size_t)arow * K + half * 2;
    const float* Wp = W + (size_t)(nt * 16 + l15) * K + half * 2;

    v8f c = {0.f, 0.f, 0.f, 0.f, 0.f, 0.f, 0.f, 0.f};
#pragma unroll 8
    for (int k = 0; k < K; k += 4) {
        v2f a = *(const v2f*)(Ap + k);
        if (relu) { a.x = fmaxf(a.x, 0.f); a.y = fmaxf(a.y, 0.f); }
        v2f b = *(const v2f*)(Wp + k);
        c = __builtin_amdgcn_wmma_f32_16x16x4_f32(false, a, false, b,
                                                  (short)0, c, false, false);
    }
    float bv = bias ? bias[nt * 16 + l15] : 0.f;
#pragma unroll
    for (int r = 0; r < 8; ++r) {
        int row = mt * 16 + half * 8 + r;
        if (row < M) C[(size_t)row * N + nt * 16 + l15] = c[r] + bv;
    }
}

// ---------------------------------------------------------------------------
// edge mapping helper: edges [0,E) from edge_index; [E, E+NENT) are the
// synthetic query->entity self edges (src = NENT + batch[i], tgt = i, rel=500)
__device__ __forceinline__ void edge_st(int e, const int* eidx, const int* batch,
                                        int& s, int& t, int& r) {
    if (e < EE) { s = eidx[e]; t = eidx[EE + e]; r = -1; }
    else        { int i = e - EE; s = NENT + batch[i]; t = i; r = RREL; }
}

// K4: alpha_raw[e][h] = sum_d lrelu(x_l[s]+x_r[t]+rel_e[r])[h][d]*att[h][d]
//     + atomic segment max per (t,h).  One wave per edge, float4 per lane.
__global__ __launch_bounds__(256)
void edge_alpha(const float* __restrict__ xl, const float* __restrict__ xr,
                const float* __restrict__ rele, const int* __restrict__ eidx,
                const int* __restrict__ ridx, const int* __restrict__ batch,
                const float* __restrict__ att, float* __restrict__ alpha,
                unsigned* __restrict__ amaxkey, int Etot) {
    int e = blockIdx.x * (blockDim.x >> 5) + (threadIdx.x >> 5);
    if (e >= Etot) return;
    int lane = threadIdx.x & 31;
    int s, t, r;
    edge_st(e, eidx, batch, s, t, r);
    if (r < 0) r = ridx[e];

    const float4* xs = (const float4*)(xl + (size_t)s * 256);
    const float4* xt = (const float4*)(xr + (size_t)t * 256);
    const float4* re = (const float4*)(rele + (size_t)r * 256);
    const float4* at = (const float4*)att;

#pragma unroll
    for (int h = 0; h < HH; ++h) {
        int idx = h * 32 + lane;                 // float4 index in 256-float row
        float4 a = xs[idx], b = xt[idx], cc = re[idx], w = at[idx];
        float v = lrelu(a.x + b.x + cc.x) * w.x
                + lrelu(a.y + b.y + cc.y) * w.y
                + lrelu(a.z + b.z + cc.z) * w.z
                + lrelu(a.w + b.w + cc.w) * w.w;
#pragma unroll
        for (int o = 16; o > 0; o >>= 1) v += __shfl_xor(v, o, 32);
        if (lane == 0) {
            alpha[(size_t)e * 2 + h] = v;
            atomicMax(&amaxkey[(size_t)t * 2 + h], fkey(v));
        }
    }
}

// K5: alpha <- exp(alpha - amax[tgt]);  denom[tgt] += alpha   (per (e,h))
__global__ void edge_exp(const int* __restrict__ eidx, const int* __restrict__ batch,
                         float* __restrict__ alpha, const unsigned* __restrict__ amaxkey,
                         float* __restrict__ denom, int Etot) {
    int i = blockIdx.x * blockDim.x + threadIdx.x;
    if (i >= Etot * 2) return;
    int e = i >> 1, h = i & 1;
    int t = (e < EE) ? eidx[EE + e] : (e - EE);
    float am = fdecode(amaxkey[(size_t)t * 2 + h]);
    float ex = expf(alpha[i] - am);
    alpha[i] = ex;
    atomicAdd(&denom[(size_t)t * 2 + h], ex);
}

// K6: out[t][d] += 0.5*(x_l[s][0][d]*w0 + x_l[s][1][d]*w1)   (head mean fused)
__global__ __launch_bounds__(256)
void edge_msg(const float* __restrict__ xl, const float* __restrict__ alpha,
              const float* __restrict__ denom, const int* __restrict__ eidx,
              const int* __restrict__ batch, float* __restrict__ out, int Etot) {
    int e = blockIdx.x * (blockDim.x >> 5) + (threadIdx.x >> 5);
    if (e >= Etot) return;
    int lane = threadIdx.x & 31;
    int s, t, r;
    edge_st(e, eidx, batch, s, t, r);
    float w0 = alpha[(size_t)e * 2 + 0] / (denom[(size_t)t * 2 + 0] + 1e-16f);
    float w1 = alpha[(size_t)e * 2 + 1] / (denom[(size_t)t * 2 + 1] + 1e-16f);
    const float4* xs = (const float4*)(xl + (size_t)s * 256);
    float4 a0 = xs[lane];        // head 0
    float4 a1 = xs[32 + lane];   // head 1
    float* op = out + (size_t)t * DD + lane * 4;
    atomicAdd(op + 0, 0.5f * (a0.x * w0 + a1.x * w1));
    atomicAdd(op + 1, 0.5f * (a0.y * w0 + a1.y * w1));
    atomicAdd(op + 2, 0.5f * (a0.z * w0 + a1.z * w1));
    atomicAdd(op + 3, 0.5f * (a0.w * w0 + a1.w * w1));
}

// K7: out[i] += bias[i % 128]
__global__ void add_bias(float* __restrict__ out, const float* __restrict__ bias,
                         int n) {
    int i = blockIdx.x * blockDim.x + threadIdx.x;
    if (i < n) out[i] += bias[i & (DD - 1)];
}

// ---------------------------------------------------------------------------
extern "C" void kernel_launch(void* const* d_in, const int* in_sizes, int n_in,
                              void* d_out, int out_size, void* d_ws, size_t ws_size,
                              hipStream_t stream) {
    const float* queries   = (const float*)d_in[0];
    const float* entities  = (const float*)d_in[1];
    const int*   edge_idx  = (const int*)d_in[2];
    const float* relations = (const float*)d_in[3];
    const int*   rel_idx   = (const int*)d_in[4];
    const int*   batch     = (const int*)d_in[5];
    const float* W_l  = (const float*)d_in[6];
    const float* b_l  = (const float*)d_in[7];
    const float* W_r  = (const float*)d_in[8];
    const float* b_r  = (const float*)d_in[9];
    const float* W_ea = (const float*)d_in[10];
    const float* att  = (const float*)d_in[11];
    const float* bias = (const float*)d_in[12];
    const float* W_le = (const float*)d_in[13];
    const float* b_le = (const float*)d_in[14];

    float* out = (float*)d_out;
    const int n_tot = NENT + BQ;       // 100064
    const int Etot  = EE + NENT;       // 500000

    // workspace layout
    float* ws = (float*)d_ws;
    size_t off = 0;
    float* xl      = ws + off; off += (size_t)n_tot * 256;
    float* xr      = ws + off; off += (size_t)n_tot * 256;
    float* relfeat = ws + off; off += (size_t)(RREL + 1) * DD;
    float* rele    = ws + off; off += (size_t)(RREL + 1) * 256;
    float* alpha   = ws + off; off += (size_t)Etot * 2;
    unsigned* amaxkey = (unsigned*)(ws + off); off += (size_t)n_tot * 2;
    float* denom   = ws + off; off += (size_t)n_tot * 2;

    const int TB = 256;
    const int nout = NENT * DD;                         // 12.8M
    // K0: init
    init_buffers<<<(nout + TB - 1) / TB, TB, 0, stream>>>(
        amaxkey, denom, out, n_tot * 2, nout);
    // K1: rel_feat
    build_relfeat<<<((RREL + 1) * DD + TB - 1) / TB, TB, 0, stream>>>(
        relations, relfeat);
    // K2: x_l / x_r (WMMA), 6254*16 waves per z
    {
        int waves = (n_tot >> 4) * 16;
        dim3 grid((waves + (TB / 32) - 1) / (TB / 32), 1, 2);
        gemm_xlr<<<grid, TB, 0, stream>>>(entities, queries, W_l, b_l, W_r, b_r,
                                          xl, xr, n_tot);
    }
    // K3: rel_e = rel_feat @ W_ea.T  (501 x 256, WMMA)
    {
        int waves = ((RREL + 1 + 15) / 16) * (256 / 16);
        gemm_generic<<<(waves + (TB / 32) - 1) / (TB / 32), TB, 0, stream>>>(
            relfeat, W_ea, nullptr, rele, RREL + 1, DD, 256, 0);
    }
    // K4: alpha + segment max  (one wave per edge)
    {
        int blocks = (Etot + (TB / 32) - 1) / (TB / 32);
        edge_alpha<<<blocks, TB, 0, stream>>>(xl, xr, rele, edge_idx, rel_idx,
                                              batch, att, alpha, amaxkey, Etot);
    }
    // K5: exp + segment sum
    edge_exp<<<(Etot * 2 + TB - 1) / TB, TB, 0, stream>>>(
        edge_idx, batch, alpha, amaxkey, denom, Etot);
    // K6: weighted scatter into out (head mean fused)
    {
        int blocks = (Etot + (TB / 32) - 1) / (TB / 32);
        edge_msg<<<blocks, TB, 0, stream>>>(xl, alpha, denom, edge_idx, batch,
                                            out, Etot);
    }
    // K7: + bias
    add_bias<<<(nout + TB - 1) / TB, TB, 0, stream>>>(out, bias, nout);
    // K8: out_edge = relu(rel_e[:500]) @ W_le.T + b_le   (WMMA, relu fused)
    {
        int waves = ((RREL + 15) / 16) * (DD / 16);
        gemm_generic<<<(waves + (TB / 32) - 1) / (TB / 32), TB, 0, stream>>>(
            rele, W_le, b_le, out + (size_t)NENT * DD, RREL, 256, DD, 1);
    }
}